// DiffeomorphicTransform_unit_19069654794929
// MI455X (gfx1250) — compile-verified
//
#include <hip/hip_runtime.h>
#include <stdint.h>

// DiffeomorphicTransform (scaling & squaring), B=1, C=3, D=H=W=128, 7 steps.
// Internal flow stored AoS float4 (fx,fy,fz,0) so each trilinear corner is ONE
// global_load_b128. Ping-pong buffers live in d_ws (needs 2*128^3*16B = 64MB),
// which fits entirely in MI455X's 192MB L2 -> L2-bandwidth bound, not HBM.

#define DIMW 128
#define NVOX (DIMW * DIMW * DIMW)   // 2097152
#define SCALE_C 63.5f               // 0.5f * (128 - 1): pos = idx + flow*SCALE_C
#define BLOCK 256                   // 8 wave32 waves per block

struct alignas(16) f4 { float x, y, z, w; };

__device__ __forceinline__ f4 lerp4(const f4 a, const f4 b, float t) {
    f4 r;
    r.x = fmaf(t, b.x - a.x, a.x);
    r.y = fmaf(t, b.y - a.y, a.y);
    r.z = fmaf(t, b.z - a.z, a.z);
    r.w = 0.0f;
    return r;
}

__device__ __forceinline__ const f4& corner(const f4* __restrict__ s, int z, int y, int x) {
    return s[(z * DIMW + y) * DIMW + x];
}

// velocity planar [3][N] -> flow AoS float4, scaled by 2^-TIME_STEP
__global__ __launch_bounds__(BLOCK)
void dt_init(const float* __restrict__ vel, f4* __restrict__ dst) {
    int i = blockIdx.x * BLOCK + threadIdx.x;
    const float s = 1.0f / 128.0f;  // 2^-7
    f4 r;
    r.x = vel[i] * s;
    r.y = vel[NVOX + i] * s;
    r.z = vel[2 * NVOX + i] * s;
    r.w = 0.0f;
    dst[i] = r;
}

// One squaring step: dst = src + trilinear_sample(src, identity + src)
// Last iteration writes planar f32 to outPlanar instead of dst.
__global__ __launch_bounds__(BLOCK)
void dt_iter(const f4* __restrict__ src, f4* __restrict__ dst,
             float* __restrict__ outPlanar, int writePlanar) {
    __shared__ f4 sh[BLOCK];
    const int i = blockIdx.x * BLOCK + threadIdx.x;

    // Stage this block's own flow row (4KB contiguous) into LDS with the
    // CDNA5 async global->LDS path (ASYNCcnt-tracked DMA, bypasses VGPRs).
    {
        unsigned lds_addr = (unsigned)(uintptr_t)(&sh[threadIdx.x]);  // flat[31:0] == LDS byte addr
        const f4* gaddr = src + i;
        asm volatile("global_load_async_to_lds_b128 %0, %1, off"
                     :: "v"(lds_addr), "v"(gaddr) : "memory");
        asm volatile("s_wait_asynccnt 0" ::: "memory");
    }
    __syncthreads();
    const f4 f = sh[threadIdx.x];

    const int wv = i & (DIMW - 1);
    const int hv = (i >> 7) & (DIMW - 1);
    const int dv = i >> 14;

    // grid_sample coords, align_corners=True: p = idx + flow * 0.5*(dim-1)
    const float px = (float)wv + f.x * SCALE_C;
    const float py = (float)hv + f.y * SCALE_C;
    const float pz = (float)dv + f.z * SCALE_C;

    const float x0f = floorf(px), y0f = floorf(py), z0f = floorf(pz);
    const float wx = px - x0f, wy = py - y0f, wz = pz - z0f;

    // border padding: clamp indices, keep weights (matches reference exactly)
    int x0 = (int)x0f, y0 = (int)y0f, z0 = (int)z0f;
    int x1 = min(max(x0 + 1, 0), DIMW - 1);
    int y1 = min(max(y0 + 1, 0), DIMW - 1);
    int z1 = min(max(z0 + 1, 0), DIMW - 1);
    x0 = min(max(x0, 0), DIMW - 1);
    y0 = min(max(y0, 0), DIMW - 1);
    z0 = min(max(z0, 0), DIMW - 1);

    // 8 corner gathers, each a single b128 load (AoS payoff)
    const f4 c000 = corner(src, z0, y0, x0), c001 = corner(src, z0, y0, x1);
    const f4 c010 = corner(src, z0, y1, x0), c011 = corner(src, z0, y1, x1);
    const f4 c100 = corner(src, z1, y0, x0), c101 = corner(src, z1, y0, x1);
    const f4 c110 = corner(src, z1, y1, x0), c111 = corner(src, z1, y1, x1);

    const f4 c00 = lerp4(c000, c001, wx);
    const f4 c01 = lerp4(c010, c011, wx);
    const f4 c10 = lerp4(c100, c101, wx);
    const f4 c11 = lerp4(c110, c111, wx);
    const f4 c0  = lerp4(c00, c01, wy);
    const f4 c1  = lerp4(c10, c11, wy);
    const f4 cc  = lerp4(c0, c1, wz);

    f4 r;
    r.x = f.x + cc.x;
    r.y = f.y + cc.y;
    r.z = f.z + cc.z;
    r.w = 0.0f;

    if (writePlanar) {
        outPlanar[i]            = r.x;
        outPlanar[NVOX + i]     = r.y;
        outPlanar[2 * NVOX + i] = r.z;
    } else {
        dst[i] = r;
    }
}

extern "C" void kernel_launch(void* const* d_in, const int* in_sizes, int n_in,
                              void* d_out, int out_size, void* d_ws, size_t ws_size,
                              hipStream_t stream) {
    (void)in_sizes; (void)n_in; (void)out_size; (void)ws_size;
    const float* velocity = (const float*)d_in[0];
    // d_in[1] (sample_grid) is the analytic identity grid; computed in-kernel.
    float* out = (float*)d_out;

    f4* bufA = (f4*)d_ws;            // NVOX float4 = 32MB
    f4* bufB = bufA + NVOX;          // +32MB (ws_size must be >= 64MB)

    const dim3 blk(BLOCK);
    const dim3 grd(NVOX / BLOCK);    // 8192 blocks

    dt_init<<<grd, blk, 0, stream>>>(velocity, bufA);

    f4* src = bufA;
    f4* dst = bufB;
    const int T = 7;                 // TIME_STEP
    for (int t = 0; t < T; ++t) {
        const int last = (t == T - 1) ? 1 : 0;
        dt_iter<<<grd, blk, 0, stream>>>(src, dst, out, last);
        f4* tmp = src; src = dst; dst = tmp;
    }
}